// OutputNetwork_67551245632087
// MI455X (gfx1250) — compile-verified
//
#include <hip/hip_runtime.h>
#include <math.h>

typedef float v2f __attribute__((ext_vector_type(2)));
typedef float v8f __attribute__((ext_vector_type(8)));

#define HCH 128      // hidden channels
#define H2  64       // H/2
#define LDSW (HCH + 2)   // padded row stride (floats) to avoid LDS bank conflicts

__device__ __forceinline__ float silu_f(float v) {
    return v / (1.0f + __expf(-v));
}

// ---------------- zero workspace ----------------
__global__ void zero_ws_kernel(float* p, int n) {
    int i = blockIdx.x * blockDim.x + threadIdx.x;
    if (i < n) p[i] = 0.0f;
}

// ---------------- per-atom mass/CoM accumulation ----------------
__global__ void mass_accum_kernel(const float* __restrict__ pos,
                                  const float* __restrict__ atomic_mass,
                                  const int* __restrict__ z,
                                  const int* __restrict__ batch,
                                  float* __restrict__ msum,   // [B][4]: m, m*x, m*y, m*z
                                  int n) {
    int i = blockIdx.x * blockDim.x + threadIdx.x;
    if (i >= n) return;
    float m = atomic_mass[z[i]];
    int b = batch[i];
    float px = pos[(size_t)3 * i + 0];
    float py = pos[(size_t)3 * i + 1];
    float pz = pos[(size_t)3 * i + 2];
    atomicAdd(&msum[(size_t)4 * b + 0], m);
    atomicAdd(&msum[(size_t)4 * b + 1], m * px);
    atomicAdd(&msum[(size_t)4 * b + 2], m * py);
    atomicAdd(&msum[(size_t)4 * b + 3], m * pz);
}

// ---------------- finalize center of mass ----------------
__global__ void com_finalize_kernel(const float* __restrict__ msum,
                                    float* __restrict__ com,  // [B][4]
                                    int B) {
    int b = blockIdx.x * blockDim.x + threadIdx.x;
    if (b >= B) return;
    float m = msum[(size_t)4 * b + 0];
    float inv = 1.0f / (m > 0.0f ? m : 1.0f);
    com[(size_t)4 * b + 0] = msum[(size_t)4 * b + 1] * inv;
    com[(size_t)4 * b + 1] = msum[(size_t)4 * b + 2] * inv;
    com[(size_t)4 * b + 2] = msum[(size_t)4 * b + 3] * inv;
}

// ---------------- fused MLP (WMMA f32) + dipole accumulation ----------------
// Each wave processes 16-atom tiles (grid-stride). Per tile:
//   C[16x64] = A[16x128] * B[128x64]  via 4 N-tiles x 32 K-steps of
//   V_WMMA_F32_16X16X4_F32, then xo = W2 . silu(C + b1) + b2 (cross-lane dot),
//   then atomic segment-add of xo*(pos - com).
__global__ __launch_bounds__(128)
void mlp_dipole_kernel(const float* __restrict__ x,      // [N][128]
                       const float* __restrict__ pos,    // [N][3]
                       const float* __restrict__ W1,     // [64][128]
                       const float* __restrict__ b1,     // [64]
                       const float* __restrict__ W2,     // [64]
                       const float* __restrict__ b2,     // [1]
                       const int*   __restrict__ batch,  // [N]
                       const float* __restrict__ com,    // [B][4]
                       float* __restrict__ acc,          // [B][4] dipole accumulators
                       int nAtoms, int nTiles, int totalWaves) {
    __shared__ float lds_w1[H2 * LDSW];
    __shared__ float lds_b1[H2];
    __shared__ float lds_w2[H2];

    const int tid = threadIdx.x;

    // Cooperative, coalesced stage of W1 into LDS (padded rows: stride 130 dwords
    // -> per-lane bank = (2n + k) & 63, conflict-free across the 16-lane group).
    for (int i = tid; i < H2 * HCH; i += blockDim.x) {
        int nrow = i >> 7;        // / 128
        int k    = i & (HCH - 1);
        lds_w1[nrow * LDSW + k] = W1[i];
    }
    if (tid < H2) {
        lds_b1[tid] = b1[tid];
        lds_w2[tid] = W2[tid];
    }
    __syncthreads();

    const float b2v   = b2[0];
    const int lane    = tid & 31;
    const int nlane   = lane & 15;   // N / M position within half-wave
    const int hi      = lane >> 4;   // 0: K pair {0,1}; 1: K pair {2,3}
    const int gwave   = blockIdx.x * 4 + (tid >> 5);

    for (int tile = gwave; tile < nTiles; tile += totalWaves) {
        const int atomBase = tile * 16;
        int myAtom = atomBase + nlane;
        int ldAtom = myAtom < nAtoms ? myAtom : (nAtoms - 1); // clamp: keep EXEC full
        const float* xrow = x + (size_t)ldAtom * HCH;

        v8f c0 = {}; v8f c1 = {}; v8f c2 = {}; v8f c3 = {};

        #pragma unroll 8
        for (int kb = 0; kb < HCH; kb += 4) {
            const int koff = kb + 2 * hi;
            // A fragment: lane (M=nlane) holds x[M][koff], x[M][koff+1]
            v2f a;
            a.x = xrow[koff];
            a.y = xrow[koff + 1];
            // B fragments: lane (N=16*t+nlane) holds W1[N][koff], W1[N][koff+1]
            const float* wr = &lds_w1[nlane * LDSW + koff];
            v2f bv;
            bv.x = wr[0];             bv.y = wr[1];
            c0 = __builtin_amdgcn_wmma_f32_16x16x4_f32(false, a, false, bv, (short)0, c0, false, false);
            wr += 16 * LDSW;
            bv.x = wr[0];             bv.y = wr[1];
            c1 = __builtin_amdgcn_wmma_f32_16x16x4_f32(false, a, false, bv, (short)0, c1, false, false);
            wr += 16 * LDSW;
            bv.x = wr[0];             bv.y = wr[1];
            c2 = __builtin_amdgcn_wmma_f32_16x16x4_f32(false, a, false, bv, (short)0, c2, false, false);
            wr += 16 * LDSW;
            bv.x = wr[0];             bv.y = wr[1];
            c3 = __builtin_amdgcn_wmma_f32_16x16x4_f32(false, a, false, bv, (short)0, c3, false, false);
        }

        // Second linear layer: xo[m] = sum_n silu(h[m][n] + b1[n]) * W2[n] + b2.
        // C layout: VGPR r -> M=r (lanes 0-15) / M=r+8 (lanes 16-31), N = nlane (+16t).
        const float b1v0 = lds_b1[ 0 + nlane], w2v0 = lds_w2[ 0 + nlane];
        const float b1v1 = lds_b1[16 + nlane], w2v1 = lds_w2[16 + nlane];
        const float b1v2 = lds_b1[32 + nlane], w2v2 = lds_w2[32 + nlane];
        const float b1v3 = lds_b1[48 + nlane], w2v3 = lds_w2[48 + nlane];

        float xoMine = 0.0f;
        #pragma unroll
        for (int r = 0; r < 8; ++r) {
            float p = silu_f(c0[r] + b1v0) * w2v0
                    + silu_f(c1[r] + b1v1) * w2v1
                    + silu_f(c2[r] + b1v2) * w2v2
                    + silu_f(c3[r] + b1v3) * w2v3;
            // reduce across the 16-lane half (xor masks < 16 stay inside halves)
            p += __shfl_xor(p, 1);
            p += __shfl_xor(p, 2);
            p += __shfl_xor(p, 4);
            p += __shfl_xor(p, 8);
            if (nlane == r) xoMine = p;   // lane r -> atom r ; lane 16+r -> atom r+8
        }

        // Lanes 0-7 own atoms 0-7, lanes 16-23 own atoms 8-15 of this tile.
        const int mLocal = hi * 8 + (nlane & 7);
        const int atomI  = atomBase + mLocal;
        if (nlane < 8 && atomI < nAtoms) {
            float xo = xoMine + b2v;
            int b = batch[atomI];
            float px = pos[(size_t)3 * atomI + 0];
            float py = pos[(size_t)3 * atomI + 1];
            float pz = pos[(size_t)3 * atomI + 2];
            const float* cb = com + (size_t)4 * b;
            atomicAdd(&acc[(size_t)4 * b + 0], xo * (px - cb[0]));
            atomicAdd(&acc[(size_t)4 * b + 1], xo * (py - cb[1]));
            atomicAdd(&acc[(size_t)4 * b + 2], xo * (pz - cb[2]));
        }
    }
}

// ---------------- per-segment L2 norm ----------------
__global__ void norm_kernel(const float* __restrict__ acc, float* __restrict__ out, int B) {
    int b = blockIdx.x * blockDim.x + threadIdx.x;
    if (b >= B) return;
    float ax = acc[(size_t)4 * b + 0];
    float ay = acc[(size_t)4 * b + 1];
    float az = acc[(size_t)4 * b + 2];
    out[b] = sqrtf(ax * ax + ay * ay + az * az);
}

extern "C" void kernel_launch(void* const* d_in, const int* in_sizes, int n_in,
                              void* d_out, int out_size, void* d_ws, size_t ws_size,
                              hipStream_t stream) {
    const float* x    = (const float*)d_in[0];
    const float* pos  = (const float*)d_in[1];
    const float* W1   = (const float*)d_in[2];
    const float* b1   = (const float*)d_in[3];
    const float* W2   = (const float*)d_in[4];
    const float* b2   = (const float*)d_in[5];
    const float* am   = (const float*)d_in[6];
    const int*   z    = (const int*)d_in[7];
    const int*   bat  = (const int*)d_in[8];
    // d_in[9] = num_segments (device scalar); B is recoverable from out_size.

    const int n = in_sizes[7];       // N atoms (size of z)
    const int B = out_size;          // segments (output is [B,1])

    float* msum = (float*)d_ws;            // [B][4]
    float* com  = msum + (size_t)B * 4;    // [B][4]
    float* acc  = com  + (size_t)B * 4;    // [B][4]
    float* out  = (float*)d_out;

    // 1) zero msum + com + acc (12*B floats)
    {
        int tot = 12 * B;
        zero_ws_kernel<<<(tot + 255) / 256, 256, 0, stream>>>(msum, tot);
    }
    // 2) mass-weighted accumulation
    mass_accum_kernel<<<(n + 255) / 256, 256, 0, stream>>>(pos, am, z, bat, msum, n);
    // 3) finalize centers of mass
    com_finalize_kernel<<<(B + 255) / 256, 256, 0, stream>>>(msum, com, B);
    // 4) fused WMMA MLP + dipole segment-sum
    {
        const int nTiles = (n + 15) / 16;
        const int blocks = 2048;                 // persistent grid-stride over tiles
        const int totalWaves = blocks * 4;       // 128 threads = 4 waves/block
        mlp_dipole_kernel<<<blocks, 128, 0, stream>>>(x, pos, W1, b1, W2, b2,
                                                      bat, com, acc,
                                                      n, nTiles, totalWaves);
    }
    // 5) per-segment norm
    norm_kernel<<<(B + 255) / 256, 256, 0, stream>>>(acc, out, B);
}